// GTStepwiseConstantVelocityModel_74612171866618
// MI455X (gfx1250) — compile-verified
//
#include <hip/hip_runtime.h>
#include <hip/hip_bf16.h>
#include <math.h>

#define N_NODES 384
#define T_STEPS 128
#define N_EVENTS 256
#define TILE 16
#define NTILE (N_NODES / TILE)                 // 24
#define NTILEPAIRS (NTILE * (NTILE + 1) / 2)   // 300 (I <= J tiles)

typedef __attribute__((ext_vector_type(2))) float v2f;
typedef __attribute__((ext_vector_type(8))) float v8f;

// Branchless erf (Abramowitz-Stegun 7.1.26, |abs err| <= 1.5e-7).
// Uses v_rcp_f32 + v_exp_f32; no divergent slow paths, EXEC stays all-1s.
__device__ __forceinline__ float erf_fast(float x) {
  float ax = fabsf(x);
  float t  = __builtin_amdgcn_rcpf(fmaf(0.3275911f, ax, 1.0f));
  float poly = t * fmaf(t, fmaf(t, fmaf(t, fmaf(t, 1.061405429f, -1.453152027f),
                                        1.421413741f), -0.284496736f),
                        0.254829592f);
  float r = 1.0f - poly * __expf(-ax * ax);
  return copysignf(r, x);
}

// ---------------------------------------------------------------------------
// Kernel A: trajectory scan.  Produces t-major arrays:
//   Zs[t][n][d] = z0 + dt * sum_{s<t} v0[n][d][s]   (positions at step starts)
//   Vr[t][n][d] = v0[n][d][t]                        (t-major velocity copy)
//   S [t][n]    = { |z|^2, |v|^2, z.v, 0 }           (per-node diagonal terms)
// Also zeroes the scalar output (runs before B/C due to stream ordering).
// ---------------------------------------------------------------------------
__global__ void precompute_kernel(const float* __restrict__ z0,
                                  const float* __restrict__ v0,
                                  const float* __restrict__ t0p,
                                  const float* __restrict__ tnp,
                                  float* __restrict__ Zs,
                                  float* __restrict__ Vr,
                                  float* __restrict__ S,
                                  float* __restrict__ out) {
  int n = blockIdx.x * blockDim.x + threadIdx.x;
  if (n == 0 && blockIdx.x == 0) out[0] = 0.0f;
  if (n >= N_NODES) return;
  const float dt = (tnp[0] - t0p[0]) / (float)T_STEPS;
  float zx = z0[n * 2 + 0];
  float zy = z0[n * 2 + 1];
  for (int t = 0; t < T_STEPS; ++t) {
    float vx = v0[(n * 2 + 0) * T_STEPS + t];   // v0 layout (N, D, T)
    float vy = v0[(n * 2 + 1) * T_STEPS + t];
    int base = t * N_NODES * 2 + n * 2;
    Zs[base + 0] = zx;  Zs[base + 1] = zy;
    Vr[base + 0] = vx;  Vr[base + 1] = vy;
    int sb = (t * N_NODES + n) * 4;
    S[sb + 0] = zx * zx + zy * zy;
    S[sb + 1] = vx * vx + vy * vy;
    S[sb + 2] = zx * vx + zy * vy;
    S[sb + 3] = 0.0f;
    zx = fmaf(vx, dt, zx);
    zy = fmaf(vy, dt, zy);
  }
}

// ---------------------------------------------------------------------------
// Kernel B: non-event integral over all i<j pairs and all T steps.
// One block = one 16x16 (I,J) tile of the pair matrix (I <= J), 4 waves,
// waves strided over t.  Per (tile, t), four V_WMMA_F32_16X16X4_F32 ops
// compute ALL pairwise reductions (diagonal terms folded into K slots, the
// mixed dot product chained through the C accumulator):
//   Dz = |z_i-z_j|^2      A=(zi0,zi1,|zi|^2,1)  B=(-2zj0,-2zj1,1,|zj|^2)
//   Dv = |v_i-v_j|^2      A=(vi0,vi1,|vi|^2,1)  B=(-2vj0,-2vj1,1,|vj|^2)
//   P  = zv_i+zv_j-zi.vj  A=(zi0,zi1,zv_i,1)    B=(-vj0,-vj1,1,zv_j)
//   BB = P - vi.zj        A=(vi0,vi1,0,0)       B=(-zj0,-zj1,0,0)  C=P
// => BB[r] = (z_i-z_j).(v_i-v_j).  No LDS/shuffle traffic in the loop;
// everything downstream is branchless VALU (rsq/exp/poly-erf), EXEC all-1s.
// ---------------------------------------------------------------------------
__global__ void __launch_bounds__(128)
pair_kernel(const float* __restrict__ Zs,
            const float* __restrict__ Vr,
            const float* __restrict__ S,
            const float* __restrict__ beta,
            const float* __restrict__ t0p,
            const float* __restrict__ tnp,
            float* __restrict__ out) {
  const float dt = (tnp[0] - t0p[0]) / (float)T_STEPS;

  // Decode blockIdx.x -> upper-triangular tile (I, J), I <= J. Uniform loop.
  int p = blockIdx.x;
  int I = 0;
  while (p >= NTILE - I) { p -= NTILE - I; ++I; }
  const int J = I + p;

  const int lane = threadIdx.x & 31;
  const int wave = threadIdx.x >> 5;
  const int c    = lane & 15;          // row/column slot this lane feeds
  const bool hi  = (lane >> 4) != 0;   // lane group: K=0,1 (lo) vs K=2,3 (hi)
  const int nI = I * TILE + c;
  const int nJ = J * TILE + c;
  const float HALF_SQRT_PI = 0.886226925452758f;   // sqrt(pi)/2

  float acc = 0.0f;

  for (int t = wave; t < T_STEPS; t += 4) {
    const float2* Zt = (const float2*)(Zs + (size_t)t * N_NODES * 2);
    const float2* Vt = (const float2*)(Vr + (size_t)t * N_NODES * 2);
    const float4* St = (const float4*)(S  + (size_t)t * N_NODES * 4);

    float2 zi = Zt[nI], vi = Vt[nI];
    float2 zj = Zt[nJ], vj = Vt[nJ];
    float4 si = St[nI], sj = St[nJ];   // {|z|^2, |v|^2, z.v, 0}

    // A 16x4: lanes 0-15 supply K0/K1 of row M=c, lanes 16-31 supply K2/K3.
    // B 4x16 mirrors this split.
    v2f a1, b1, a2v, b2, a3, b3, a4, b4;
    a1.x  = hi ? si.x : zi.x;          a1.y  = hi ? 1.0f  : zi.y;
    b1.x  = hi ? 1.0f : -2.0f * zj.x;  b1.y  = hi ? sj.x  : -2.0f * zj.y;
    a2v.x = hi ? si.y : vi.x;          a2v.y = hi ? 1.0f  : vi.y;
    b2.x  = hi ? 1.0f : -2.0f * vj.x;  b2.y  = hi ? sj.y  : -2.0f * vj.y;
    a3.x  = hi ? si.z : zi.x;          a3.y  = hi ? 1.0f  : zi.y;
    b3.x  = hi ? 1.0f : -vj.x;         b3.y  = hi ? sj.z  : -vj.y;
    a4.x  = hi ? 0.0f : vi.x;          a4.y  = hi ? 0.0f  : vi.y;
    b4.x  = hi ? 0.0f : -zj.x;         b4.y  = hi ? 0.0f  : -zj.y;

    v8f z8 = {};
    v8f Dz = __builtin_amdgcn_wmma_f32_16x16x4_f32(false, a1, false, b1,
                                                   (short)0, z8, false, false);
    v8f Dv = __builtin_amdgcn_wmma_f32_16x16x4_f32(false, a2v, false, b2,
                                                   (short)0, z8, false, false);
    v8f Pm = __builtin_amdgcn_wmma_f32_16x16x4_f32(false, a3, false, b3,
                                                   (short)0, z8, false, false);
    v8f BB = __builtin_amdgcn_wmma_f32_16x16x4_f32(false, a4, false, b4,
                                                   (short)0, Pm, false, false);
    const float bt = beta[t];

#pragma unroll
    for (int r = 0; r < 8; ++r) {
      const int m = r + (hi ? 8 : 0);            // row M of this D element
      float dzsq = Dz[r];                        // |z_i - z_j|^2
      float a2 = fmaxf(Dv[r], 1e-12f);           // max(|v_i - v_j|^2, 1e-12)
      float bb = BB[r];                          // (z_i-z_j).(v_i-v_j)
      float inva = __builtin_amdgcn_rsqf(a2);    // 1/a
      float a    = a2 * inva;                    // sqrt(a2)
      float x0   = bb * inva;                    // a*c = b/a
      float x1   = fmaf(a, dt, x0);              // a*(dt + c)
      float mm   = dzsq - x0 * x0;               // m = dzsq - b^2/a2
      float val  = __expf(bt - mm) * HALF_SQRT_PI * inva *
                   (erf_fast(x1) - erf_fast(x0));
      int gi = I * TILE + m;
      int gj = J * TILE + c;
      acc += (gi < gj) ? val : 0.0f;             // strict upper triangle
    }
  }

  // Wave reduction, then subtract (non_event enters with minus sign).
  for (int off = 16; off > 0; off >>= 1) acc += __shfl_down(acc, off, 32);
  if (lane == 0) atomicAdd(out, -acc);
}

// ---------------------------------------------------------------------------
// Kernel C: event intensity term.  E=256 events, one block.
// ---------------------------------------------------------------------------
__global__ void event_kernel(const float* __restrict__ data,
                             const float* __restrict__ beta,
                             const float* __restrict__ Zs,
                             const float* __restrict__ Vr,
                             const float* __restrict__ t0p,
                             const float* __restrict__ tnp,
                             float* __restrict__ out) {
  const float dt = (tnp[0] - t0p[0]) / (float)T_STEPS;
  int e = threadIdx.x;
  float val = 0.0f;
  if (e < N_EVENTS) {
    int   i  = (int)floorf(data[e * 3 + 0]);
    int   j  = (int)floorf(data[e * 3 + 1]);
    float te = data[e * 3 + 2];
    float idx_f = floorf(te / dt);
    int   idx = (int)(idx_f < (float)T_STEPS ? idx_f : idx_f - 1.0f);
    float rem = te - idx_f * dt;     // uses unclamped idx_f, as in reference
    int bi = idx * N_NODES * 2 + i * 2;
    int bj = idx * N_NODES * 2 + j * 2;
    float zix = fmaf(Vr[bi + 0], rem, Zs[bi + 0]);
    float ziy = fmaf(Vr[bi + 1], rem, Zs[bi + 1]);
    float zjx = fmaf(Vr[bj + 0], rem, Zs[bj + 0]);
    float zjy = fmaf(Vr[bj + 1], rem, Zs[bj + 1]);
    float dx = zix - zjx, dy = ziy - zjy;
    val = beta[idx] - (dx * dx + dy * dy);
  }
  // Block reduce (8 waves of 32).
  __shared__ float red[8];
  for (int off = 16; off > 0; off >>= 1) val += __shfl_down(val, off, 32);
  int lane = threadIdx.x & 31, w = threadIdx.x >> 5;
  if (lane == 0) red[w] = val;
  __syncthreads();
  if (threadIdx.x == 0) {
    float s = 0.0f;
    for (int k = 0; k < 8; ++k) s += red[k];
    atomicAdd(out, s);
  }
}

// ---------------------------------------------------------------------------
extern "C" void kernel_launch(void* const* d_in, const int* in_sizes, int n_in,
                              void* d_out, int out_size, void* d_ws, size_t ws_size,
                              hipStream_t stream) {
  const float* data = (const float*)d_in[0];   // (E,3) [i, j, t]
  const float* t0p  = (const float*)d_in[1];   // scalar
  const float* tnp  = (const float*)d_in[2];   // scalar
  const float* z0   = (const float*)d_in[3];   // (N,D)
  const float* v0   = (const float*)d_in[4];   // (N,D,T)
  const float* beta = (const float*)d_in[5];   // (T,)
  float* out = (float*)d_out;

  // Workspace layout (floats): Zs | Vr | S   (~1.6 MB total)
  float* Zs = (float*)d_ws;
  float* Vr = Zs + (size_t)T_STEPS * N_NODES * 2;
  float* S  = Vr + (size_t)T_STEPS * N_NODES * 2;

  precompute_kernel<<<(N_NODES + 127) / 128, 128, 0, stream>>>(
      z0, v0, t0p, tnp, Zs, Vr, S, out);
  pair_kernel<<<NTILEPAIRS, 128, 0, stream>>>(Zs, Vr, S, beta, t0p, tnp, out);
  event_kernel<<<1, 256, 0, stream>>>(data, beta, Zs, Vr, t0p, tnp, out);
}